// PEAQ_6124623364269
// MI455X (gfx1250) — compile-verified
//
#include <hip/hip_runtime.h>
#include <math.h>

typedef float v2f __attribute__((ext_vector_type(2)));
typedef float v8f __attribute__((ext_vector_type(8)));

#define NFFT_  1881
#define STEP_  941
#define NFR    468          // frames per batch row
#define NB_    8            // 2*B (pred batches 0..3, targ batches 4..7)
#define NROWS  (NB_*NFR)    // 3744 total frames (multiple of 16)
#define Z_     109
#define ZP     112          // Z padded to multiple of 16
#define NBIN   765          // spectral bins IMIN..IMAX-1
#define KP     1920         // DFT K padded (>=1881, multiple of 64)
#define NC     1536         // DFT output cols: 768 cos + 768 sin (765 used each)
#define T_     441000
#define TP_    441408       // padded signal length (>= 467*941+1919, mult of 64)
#define IMIN_  3
#define TWOPI  6.2831853071795864f
#define L2_10  3.3219280948873623f   // log2(10)

// ---- workspace layout (in floats) ----
#define OFF_BASIS 0ul
#define OFF_C     (OFF_BASIS + (size_t)KP*NC)          // DFT output [NROWS x NC]
#define OFF_BM    (OFF_C + (size_t)NROWS*NC)           // bark matrix [768 x ZP]
#define OFF_UEP   (OFF_BM + 768ul*ZP)                  // unsmeared excitation [NROWS x ZP]
#define OFF_E2    (OFF_UEP + (size_t)NROWS*ZP)         // smeared [NB_ x Z_ x NFR]
#define OFF_XP    (OFF_E2 + (size_t)NB_*Z_*NFR)        // zero-padded signal [NB_ x TP_]
#define OFF_FC    (OFF_XP + (size_t)NB_*TP_)
#define OFF_SU0   (OFF_FC  + 128)
#define OFF_WN    (OFF_SU0 + 128)
#define OFF_AI    (OFF_WN  + 128)
#define OFF_NINV  (OFF_AI  + 128)
#define OFF_FMAX  (OFF_NINV + 128)
#define OFF_FAC   (OFF_FMAX + 16)

// CDNA5 async global->LDS staging (ASYNCcnt path), with graceful fallback
#if defined(__has_builtin)
#if __has_builtin(__builtin_amdgcn_global_load_async_to_lds_b32) && \
    __has_builtin(__builtin_amdgcn_s_wait_asynccnt)
#define USE_ASYNC_LDS 1
#endif
#endif
#ifndef USE_ASYNC_LDS
#define USE_ASYNC_LDS 0
#endif

#if USE_ASYNC_LDS
__device__ __forceinline__ void async_g2l_b32(const float* g, float* l) {
    __builtin_amdgcn_global_load_async_to_lds_b32(
        (__attribute__((address_space(1))) int*)(g),
        (__attribute__((address_space(3))) int*)(l), 0, 0);
}
#endif

__device__ __forceinline__ float e10f(float x) { return exp2f(x * L2_10); }
__device__ __forceinline__ v8f vz8() { v8f v = {0.f,0.f,0.f,0.f,0.f,0.f,0.f,0.f}; return v; }

// ------------------------------------------------------------------
// per-band constants: fc, Su0, Wn (internal noise), IIR pole a, norm_inv
// ------------------------------------------------------------------
__global__ __launch_bounds__(128) void peaq_const(float* ws) {
    __shared__ float su0s[Z_], ss[Z_];
    int tid = threadIdx.x;
    const float x0 = 80.0f / 650.0f;
    const float zL = 7.0f * logf(x0 + sqrtf(x0 * x0 + 1.0f));   // 7*asinh(80/650)
    if (tid < Z_) {
        float fc  = 650.0f * sinhf((zL + 0.25f * (float)tid + 0.125f) / 7.0f);
        float su0 = -24.0f - 230.0f / fc;
        ws[OFF_FC  + tid] = fc;
        ws[OFF_SU0 + tid] = su0;
        ws[OFF_WN  + tid] = e10f(0.4f * 0.364f * powf(fc * 0.001f, -0.8f));
        float tau = 0.008f + 2.2f / fc;
        ws[OFF_AI + tid] = expf(-4.0f / (187.5f * tau));
        su0s[tid] = su0;
    }
    __syncthreads();
    if (tid < Z_) {                 // s[c] = sum_c2 Ec[c,c2]  (reference's broadcast quirk)
        int c = tid; float s = 0.f;
        for (int c2 = 0; c2 < Z_; ++c2) {
            float sl = (c <= c2) ? 27.0f : su0s[c];
            s += e10f(fminf(0.025f * (float)(c - c2) * sl, 30.0f));
        }
        ss[c] = s;
    }
    __syncthreads();
    if (tid < Z_) {                 // ninv[r] = (sum_c (Ec[r,c]/s[c])^0.4)^-2.5
        int r = tid; float t = 0.f;
        for (int c = 0; c < Z_; ++c) {
            float sl = (r <= c) ? 27.0f : su0s[r];
            float ec = e10f(fminf(0.025f * (float)(r - c) * sl, 30.0f)) / ss[c];
            t += powf(ec, 0.4f);
        }
        ws[OFF_NINV + r] = powf(t, -2.5f);
    }
}

// ------------------------------------------------------------------
// normalization factor: 10 frames of a 1019.5 Hz sine, periodic Hann,
// max |rfft| per frame (direct DFT, exact integer phase reduction)
// ------------------------------------------------------------------
__global__ __launch_bounds__(256) void peaq_fac(float* ws) {
    __shared__ float red[256];
    int i = blockIdx.x, tid = threadIdx.x;
    float mx = 0.f;
    for (int k = tid; k < 941; k += 256) {
        float re = 0.f, im = 0.f;
        for (int n = 0; n < NFFT_; ++n) {
            unsigned u = (unsigned)(i * STEP_ + n);
            unsigned long long m2 = (2039ull * (unsigned long long)u) % 88200ull; // 1019.5*u mod 44100
            float sw = __sinf(TWOPI * (float)m2 / 88200.0f);
            float wp = 0.5f * (1.0f - __cosf(TWOPI * (float)n / 1881.0f));        // periodic hann
            int mk = (int)(((long long)k * (long long)n) % NFFT_);
            float th = TWOPI * (float)mk / (float)NFFT_;
            float sn, cs; __sincosf(th, &sn, &cs);
            float v = wp * sw;
            re += v * cs; im += v * sn;
        }
        mx = fmaxf(mx, sqrtf(re * re + im * im));
    }
    red[tid] = mx; __syncthreads();
    for (int s = 128; s > 0; s >>= 1) {
        if (tid < s) red[tid] = fmaxf(red[tid], red[tid + s]);
        __syncthreads();
    }
    if (tid == 0) ws[OFF_FMAX + i] = red[0];
}

__global__ void peaq_facred(float* ws) {
    if (threadIdx.x == 0 && blockIdx.x == 0) {
        float nf = 0.f;
        for (int i = 0; i < 10; ++i) nf += ws[OFF_FMAX + i];
        nf *= 0.1f;
        ws[OFF_FAC] = powf(10.0f, 4.6f) / nf;       // 10^(92/20)/nf
    }
}

// zero-padded signal copy: xp[bb][s] (tail zeros make all GEMM gathers valid)
__global__ __launch_bounds__(256) void peaq_pad(const float* __restrict__ pred,
                                                const float* __restrict__ targ,
                                                float* __restrict__ ws) {
    long idx = (long)blockIdx.x * 256 + threadIdx.x;
    if (idx >= (long)NB_ * TP_) return;
    int bb = (int)(idx / TP_);
    int s  = (int)(idx - (long)bb * TP_);
    float v = 0.f;
    if (s < T_) v = (bb < 4) ? pred[(size_t)bb * T_ + s] : targ[(size_t)(bb - 4) * T_ + s];
    ws[OFF_XP + (size_t)idx] = v;
}

// ------------------------------------------------------------------
// DFT basis [KP x NC]: window, FAC/N norm and outer/middle-ear weight folded in
// ------------------------------------------------------------------
__global__ __launch_bounds__(256) void peaq_basis(float* ws) {
    long idx = (long)blockIdx.x * 256 + threadIdx.x;
    if (idx >= (long)KP * NC) return;
    int n = (int)(idx / NC), col = (int)(idx % NC);
    int isin = col >= 768;
    int j = isin ? col - 768 : col;
    float val = 0.f;
    if (n < NFFT_ && j < NBIN) {
        int k = j + IMIN_;
        float fac = ws[OFF_FAC];
        float w = 0.5f * (1.0f - cosf(TWOPI * (float)n / 1880.0f));   // non-periodic hann
        float fk = (22050.0f * (float)k / 941.0f) * 0.001f;            // EARW freq grid
        float W = -0.6f * 3.64f * powf(fk, -0.8f)
                + 6.5f * expf(-0.6f * (fk - 3.3f) * (fk - 3.3f))
                - 0.001f * powf(fk, 3.6f);
        float earw = e10f(W * 0.05f);
        int mk = (int)(((long long)k * (long long)n) % NFFT_);
        float th = TWOPI * (float)mk / (float)NFFT_;
        float s, c; sincosf(th, &s, &c);
        val = w * (isin ? s : c) * earw * (fac / (float)NFFT_);
    }
    ws[OFF_BASIS + (size_t)idx] = val;
}

// bark overlap matrix [768 x ZP] (rows 765..767 and cols 109..111 zero)
__global__ __launch_bounds__(256) void peaq_barkmat(float* ws) {
    int idx = blockIdx.x * 256 + threadIdx.x;
    if (idx >= 768 * ZP) return;
    int j = idx / ZP, z = idx % ZP;
    float v = 0.f;
    if (j < NBIN && z < Z_) {
        const float df = 44100.0f / 1881.0f;
        const float x0 = 80.0f / 650.0f;
        const float zL = 7.0f * logf(x0 + sqrtf(x0 * x0 + 1.0f));
        int k = j + IMIN_;
        float blo = (float)k * df - 0.5f * df, bhi = blo + df;
        float zlo = zL + 0.25f * (float)z;
        float felo = 650.0f * sinhf(zlo / 7.0f);
        float fehi = 650.0f * sinhf((zlo + 0.25f) / 7.0f);
        v = fmaxf(fminf(bhi, fehi) - fmaxf(blo, felo), 0.0f) / df;
    }
    ws[OFF_BM + idx] = v;
}

// ------------------------------------------------------------------
// GEMM1: C[3744 x 1536] = frames[3744 x 1920] * basis[1920 x 1536]
// frames gathered from zero-padded signal (window folded into basis).
// 256 thr = 8 waves; each wave 2 N-tiles -> 16x256 block tile.
// A tile staged to LDS via CDNA5 async global->LDS loads (ASYNCcnt).
// ------------------------------------------------------------------
__global__ __launch_bounds__(256) void peaq_dftgemm(float* __restrict__ ws) {
    const float* __restrict__ basis = ws + OFF_BASIS;
    const float* __restrict__ xp    = ws + OFF_XP;
    float* __restrict__ C = ws + OFF_C;
    __shared__ float As[16][68];
    int tid = threadIdx.x, lane = tid & 31, wave = tid >> 5;
    int mtile = blockIdx.x;
    int nbase = blockIdx.y * 256 + wave * 32;
    int mfrag = lane & 15, klo = (lane >> 4) * 2, nfrag = lane & 15;
    v8f acc0 = vz8(), acc1 = vz8();
    int grow0 = mtile * 16;

    for (int kc = 0; kc < KP; kc += 64) {
#pragma unroll
        for (int i = 0; i < 4; ++i) {               // stage A tile 16x64 (all addrs valid)
            int idx = tid + i * 256;
            int am = idx >> 6, ak = idx & 63;
            int grow = grow0 + am;
            int bb = grow / NFR;
            int f = grow - bb * NFR;
            const float* src = xp + (size_t)bb * TP_ + (size_t)f * STEP_ + kc + ak;
#if USE_ASYNC_LDS
            async_g2l_b32(src, &As[am][ak]);
#else
            As[am][ak] = *src;
#endif
        }
#if USE_ASYNC_LDS
        __builtin_amdgcn_s_wait_asynccnt(0);
#endif
        __syncthreads();
        int kn = (kc + 64 < KP) ? kc + 64 : kc;      // hint next B chunk into cache
        __builtin_prefetch(&basis[(size_t)kn * NC + nbase + nfrag], 0, 1);
#pragma unroll
        for (int k0 = 0; k0 < 64; k0 += 4) {
            v2f a; a.x = As[mfrag][k0 + klo]; a.y = As[mfrag][k0 + klo + 1];
            const float* bp = basis + (size_t)(kc + k0 + klo) * NC + nbase + nfrag;
            v2f b0; b0.x = bp[0];  b0.y = bp[NC];
            v2f b1; b1.x = bp[16]; b1.y = bp[NC + 16];
            acc0 = __builtin_amdgcn_wmma_f32_16x16x4_f32(false, a, false, b0, (short)0, acc0, false, false);
            acc1 = __builtin_amdgcn_wmma_f32_16x16x4_f32(false, a, false, b1, (short)0, acc1, false, false);
        }
        __syncthreads();
    }
    int mrow0 = mtile * 16 + ((lane >> 4) << 3);    // C layout: VGPR g -> row g (+8 for hi half)
#pragma unroll
    for (int g = 0; g < 8; ++g) {
        size_t o = (size_t)(mrow0 + g) * NC + nbase + nfrag;
        C[o]      = acc0[g];
        C[o + 16] = acc1[g];
    }
}

// ------------------------------------------------------------------
// GEMM2: uep[3744 x 109] = clip((re^2+im^2)[3744 x 768] * BM[768 x ZP], 1e-12) + Wn
// 224 thr = 7 waves, one 16-wide N-tile per wave (7*16 = 112 = ZP)
// ------------------------------------------------------------------
__global__ __launch_bounds__(224) void peaq_barkgemm(float* __restrict__ ws) {
    const float* __restrict__ C  = ws + OFF_C;
    const float* __restrict__ BM = ws + OFF_BM;
    float* __restrict__ uep = ws + OFF_UEP;
    int tid = threadIdx.x, lane = tid & 31, wave = tid >> 5;
    int mtile = blockIdx.x;
    int mfrag = lane & 15, klo = (lane >> 4) * 2, nfrag = lane & 15;
    int row  = mtile * 16 + mfrag;
    int ncol = wave * 16 + nfrag;
    const float* crow = C + (size_t)row * NC;
    v8f acc = vz8();
    for (int k0 = 0; k0 < 768; k0 += 4) {
        int k = k0 + klo;
        float r0 = crow[k],     i0 = crow[768 + k];
        float r1 = crow[k + 1], i1 = crow[769 + k];
        v2f a; a.x = r0 * r0 + i0 * i0; a.y = r1 * r1 + i1 * i1;
        v2f b; b.x = BM[(size_t)k * ZP + ncol]; b.y = BM[(size_t)(k + 1) * ZP + ncol];
        acc = __builtin_amdgcn_wmma_f32_16x16x4_f32(false, a, false, b, (short)0, acc, false, false);
    }
    if (ncol < Z_) {
        float wn = ws[OFF_WN + ncol];
        int mrow0 = mtile * 16 + ((lane >> 4) << 3);
#pragma unroll
        for (int g = 0; g < 8; ++g)
            uep[(size_t)(mrow0 + g) * ZP + ncol] = fmaxf(acc[g], 1e-12f) + wn;
    }
}

// ------------------------------------------------------------------
// level-dependent frequency smearing; one block per frame, thread per band.
// Uses 10^min(L/10,CLIP)==uep and normalization algebra:
// E2[c] = (sum_r El^0.4)^2.5 / (sum_r El) * ninv[c], all in log2 domain.
// Writes transposed [b, z, t].
// ------------------------------------------------------------------
__global__ __launch_bounds__(128) void peaq_smear(float* __restrict__ ws) {
    __shared__ float sus[Z_], l2u[Z_];
    int tid = threadIdx.x, row = blockIdx.x;
    const float* uep = ws + OFF_UEP + (size_t)row * ZP;
    if (tid < Z_) {
        float u = uep[tid];
        float L = 10.0f * log10f(u);
        sus[tid] = ws[OFF_SU0 + tid] + 0.2f * L;
        l2u[tid] = log2f(u);
    }
    __syncthreads();
    if (tid < Z_) {
        int c = tid;
        float S = 0.f, A = 0.f;
        for (int r = 0; r < Z_; ++r) {
            float sl = (r <= c) ? 27.0f : sus[r];
            float e  = fminf(0.025f * (float)(r - c) * sl, 30.0f);
            float lv = e * L2_10 + l2u[r];
            S += exp2f(lv);
            A += exp2f(0.4f * lv);
        }
        float E2 = powf(A, 2.5f) / S * ws[OFF_NINV + c];
        int bb = row / NFR, f = row - bb * NFR;
        ws[OFF_E2 + ((size_t)bb * Z_ + c) * NFR + f] = E2;
    }
}

// temporal IIR smearing + max(ef, e2); one thread per (batch, band) chain
__global__ __launch_bounds__(64) void peaq_iir(float* __restrict__ ws, float* __restrict__ out) {
    int id = blockIdx.x * 64 + threadIdx.x;
    if (id >= NB_ * Z_) return;
    int bb = id / Z_, z = id - bb * Z_;
    const float* e2 = ws + OFF_E2 + ((size_t)bb * Z_ + z) * NFR;
    float* o = out + ((size_t)bb * Z_ + z) * NFR;
    float a = ws[OFF_AI + z], b0 = 1.0f - a, y = 0.f;
    for (int t = 0; t < NFR; ++t) {
        float xv = e2[t];
        y = b0 * ((t == 0) ? 0.f : xv) + a * y;
        o[t] = fmaxf(y, xv);
    }
}

extern "C" void kernel_launch(void* const* d_in, const int* in_sizes, int n_in,
                              void* d_out, int out_size, void* d_ws, size_t ws_size,
                              hipStream_t stream) {
    (void)in_sizes; (void)n_in; (void)out_size; (void)ws_size;
    const float* pred = (const float*)d_in[0];
    const float* targ = (const float*)d_in[1];
    float* ws  = (float*)d_ws;
    float* out = (float*)d_out;

    peaq_const<<<1, 128, 0, stream>>>(ws);
    peaq_fac<<<10, 256, 0, stream>>>(ws);
    peaq_facred<<<1, 1, 0, stream>>>(ws);

    long npad = ((long)NB_ * TP_ + 255) / 256;
    peaq_pad<<<dim3((unsigned)npad), 256, 0, stream>>>(pred, targ, ws);

    long nbasis = ((long)KP * NC + 255) / 256;
    peaq_basis<<<dim3((unsigned)nbasis), 256, 0, stream>>>(ws);
    peaq_barkmat<<<(768 * ZP + 255) / 256, 256, 0, stream>>>(ws);

    peaq_dftgemm<<<dim3(NROWS / 16, NC / 256), 256, 0, stream>>>(ws);
    peaq_barkgemm<<<NROWS / 16, 224, 0, stream>>>(ws);
    peaq_smear<<<NROWS, 128, 0, stream>>>(ws);
    peaq_iir<<<(NB_ * Z_ + 63) / 64, 64, 0, stream>>>(ws, out);
}